// Warping_78013785964550
// MI455X (gfx1250) — compile-verified
//
#include <hip/hip_runtime.h>
#include <hip/hip_bf16.h>

// dense_image_warp, B=8 H=512 W=512 C=16, fp32.
// Memory-bound gather: ~285 MB HBM traffic -> ~12 us at 23.3 TB/s on MI455X.
//
// Strategy (roofline-driven, AI ~ 1 flop/byte -> pure data movement):
//  * one thread per pixel, 16x global_load_b128 corner reads; image loads stay
//    RT so the 134 MB image remains resident in the 192 MB global L2
//  * WGP-scope global_prefetch_b8 of the identity-location line (flow ~ N(0,1)
//    so the gather lands on/near it) to warm WGP$/L0, not just GL2
//  * TH_LOAD_NT for the read-once flow, TH_STORE_NT b128 for the write-once
//    output so neither displaces the image working set
//  * no WMMA / TDM by design: gathers are per-lane data-dependent (TDM D# is
//    wave-uniform) and there is no reduction dimension for matrix cores.

typedef float v4f __attribute__((ext_vector_type(4)));
typedef float v2f __attribute__((ext_vector_type(2)));

namespace {
constexpr int kH = 512;
constexpr int kW = 512;
constexpr int kC = 16;
constexpr int kLogW = 9;        // log2(W)
constexpr int kLogHW = 18;      // log2(H*W)
}  // namespace

__global__ __launch_bounds__(256) void dense_image_warp_kernel(
    const float* __restrict__ image,   // [B,H,W,C]
    const float* __restrict__ flow,    // [B,H,W,2]
    float* __restrict__ out,           // [B,H,W,C]
    int n_pix) {
  const int idx = blockIdx.x * blockDim.x + threadIdx.x;
  if (idx >= n_pix) return;

  const int x = idx & (kW - 1);
  const int y = (idx >> kLogW) & (kH - 1);
  const int b = idx >> kLogHW;

  // WGP-scope (SCOPE_CU == 0, default) prefetch: fills all cache levels
  // on a miss. The builtin only reaches SCOPE_SE (GL2-only), which is pointless
  // here since the image is already L2-resident; we want the WGP$ warmed.
  // Address is always valid (identity pixel), so non-speculative is safe.
  const float* id_line = image + (size_t)idx * kC;
  asm volatile("global_prefetch_b8 %0, off" : : "v"(id_line));

  // Flow is streamed exactly once: non-temporal so it does not displace the
  // image working set from L2.
  const v2f fl =
      __builtin_nontemporal_load(reinterpret_cast<const v2f*>(flow) + idx);

  const float qy = (float)y - fl.x;
  const float qx = (float)x - fl.y;

  float fy = floorf(qy);
  float fx = floorf(qx);
  fy = fminf(fmaxf(fy, 0.0f), (float)(kH - 2));
  fx = fminf(fmaxf(fx, 0.0f), (float)(kW - 2));
  const float ay = fminf(fmaxf(qy - fy, 0.0f), 1.0f);
  const float ax = fminf(fmaxf(qx - fx, 0.0f), 1.0f);
  const int iy = (int)fy;
  const int ix = (int)fx;

  // Base of top-left corner's channel vector; C=16 fp32 -> 64B aligned, so
  // float4 (b128) accesses are always naturally aligned.
  const size_t base = ((((size_t)b * kH + (size_t)iy) * kW) + (size_t)ix) * kC;
  const v4f* __restrict__ top = reinterpret_cast<const v4f*>(image + base);
  const v4f* __restrict__ bot =
      reinterpret_cast<const v4f*>(image + base + (size_t)kW * kC);

  // Issue all 16 b128 corner loads before any use: deep MLP per wave.
  v4f tl[4], tr[4], bl[4], br[4];
#pragma unroll
  for (int k = 0; k < 4; ++k) {
    tl[k] = top[k];
    tr[k] = top[4 + k];   // (ix+1) is +C floats = +4 v4f
    bl[k] = bot[k];
    br[k] = bot[4 + k];
  }

  v4f* __restrict__ o = reinterpret_cast<v4f*>(out + (size_t)idx * kC);
#pragma unroll
  for (int k = 0; k < 4; ++k) {
    const v4f t = tl[k] + ax * (tr[k] - tl[k]);   // interp_top
    const v4f bo = bl[k] + ax * (br[k] - bl[k]);  // interp_bottom
    const v4f r = t + ay * (bo - t);
    // Output is written exactly once and never re-read: NT b128 store.
    __builtin_nontemporal_store(r, o + k);
  }
}

extern "C" void kernel_launch(void* const* d_in, const int* in_sizes, int n_in,
                              void* d_out, int out_size, void* d_ws,
                              size_t ws_size, hipStream_t stream) {
  const float* image = reinterpret_cast<const float*>(d_in[0]);
  const float* flow = reinterpret_cast<const float*>(d_in[1]);
  float* out = reinterpret_cast<float*>(d_out);

  const int n_pix = in_sizes[1] / 2;  // B*H*W from flow element count
  const int block = 256;              // 8 wave32 per workgroup
  const int grid = (n_pix + block - 1) / block;
  dense_image_warp_kernel<<<grid, block, 0, stream>>>(image, flow, out, n_pix);
}